// Head_63101659513008
// MI455X (gfx1250) — compile-verified
//
#include <hip/hip_runtime.h>

// Head attention collapsed to a per-batch Gram matrix Z = Y^T Y computed with
// V_WMMA_F32_16X16X4_F32 (exact f32).
//
// Y (512 x 16) per batch: cols 0-3 = x0[m,f] = xr[b, m, f]
//                         cols 4-7 = x1[m,f] = xr[b, 512+m, f]
//                         col  8   = 1.0  (yields column sums S0/S1)
//                         cols 9-15 = 0
// Z[4+f, f']  = G10,  Z[4+f, 4+f'] = G11,  Z[8, f] = S0,  Z[8, 4+f] = S1.
//
// Memory-bound: 16 MB of x read exactly once -> ~0.7 us at 23.3 TB/s.
// All loads are unconditional (dummy in-bounds address for filler lanes +
// v_cndmask) with one base pointer and immediate offsets, so the compiler can
// keep the whole tile's loads in flight ahead of the WMMA accumulation chain.

typedef float v2f __attribute__((ext_vector_type(2)));
typedef float v8f __attribute__((ext_vector_type(8)));

#define NHEAD 10

static __device__ inline v8f wmma_acc(v2f ab, v8f c) {
    return __builtin_amdgcn_wmma_f32_16x16x4_f32(
        /*neg_a=*/false, ab, /*neg_b=*/false, ab,
        /*c_mod=*/(short)0, c, /*reuse_a=*/false, /*reuse_b=*/false);
}

// Fully unrolled Gram accumulation: ITERS chunks of K=4 rows.
// p points at this lane's first element (or xb for filler lanes).
template <int ITERS>
static __device__ inline v8f gram_accum(const float* __restrict__ p,
                                        bool is_x, float fillv) {
    v8f a0 = {0.f, 0.f, 0.f, 0.f, 0.f, 0.f, 0.f, 0.f};
    v8f a1 = {0.f, 0.f, 0.f, 0.f, 0.f, 0.f, 0.f, 0.f};
    #pragma unroll
    for (int it = 0; it < ITERS; it += 2) {
        // chunk it   -> rows m0, m0+1 for this lane-half (offsets it*16, +4)
        // chunk it+1 -> next 4 rows (offsets +16, +20)
        float u0 = p[it * 16 + 0];
        float u1 = p[it * 16 + 4];
        float w0 = p[it * 16 + 16];
        float w1 = p[it * 16 + 20];
        v2f ab0, ab1;
        ab0.x = is_x ? u0 : fillv;
        ab0.y = is_x ? u1 : fillv;
        ab1.x = is_x ? w0 : fillv;
        ab1.y = is_x ? w1 : fillv;
        a0 = wmma_acc(ab0, a0);   // independent even/odd accumulators halve
        a1 = wmma_acc(ab1, a1);   // the WMMA RAW dependency chain
    }
    return a0 + a1;
}

__global__ __launch_bounds__(256) void head_attn_gram_wmma(
    const float* __restrict__ x,   // (B, 2B, 2) == xr (B, B, 4)
    const float* __restrict__ Wk,  // (10, 4, 4)
    const float* __restrict__ Wq,  // (10, 4, 4)
    const float* __restrict__ Wv,  // (10, 4, 4)
    float* __restrict__ out,       // (B, 2, 10)
    int B)
{
    __shared__ float lds_part[8 * 256];
    __shared__ float Z[16 * 16];

    const int b    = blockIdx.x;
    const int tid  = threadIdx.x;
    const int wave = tid >> 5;
    const int lane = tid & 31;
    const int c    = lane & 15;   // Y column this lane owns
    const int half = lane >> 4;   // K-halves 0-1 vs 2-3 within a chunk

    const int halfRows     = B >> 1;   // 512
    const int itersPerWave = B >> 6;   // 16 for B=1024

    const float* __restrict__ xb = x + (size_t)b * (size_t)B * 4;

    // Per-lane element source for Y[m, c]:
    //   c in 0..3 -> xr[b, m, c];  c in 4..7 -> xr[b, halfRows+m, c-4]
    //   c == 8    -> 1.0;          c in 9..15 -> 0.0
    const bool  is_x  = (c < 8);
    const int   rowOff = (c < 4) ? 0 : halfRows;
    const int   colOff = c & 3;
    const float fillv  = (c == 8) ? 1.0f : 0.0f;

    const int   mBase  = wave * (itersPerWave * 4);
    // Filler lanes point at xb[0]: always in-bounds, hot in cache, discarded.
    const int   laneElem = is_x ? ((rowOff + mBase + half * 2) * 4 + colOff) : 0;
    const float* __restrict__ p = xb + laneElem;

    v8f acc;
    if (itersPerWave == 16) {
        acc = gram_accum<16>(p, is_x, fillv);        // hot path, B = 1024
    } else {
        acc = (v8f){0.f, 0.f, 0.f, 0.f, 0.f, 0.f, 0.f, 0.f};
        for (int it = 0; it < itersPerWave; ++it) {
            v2f ab;
            float u0 = p[it * 16 + 0];
            float u1 = p[it * 16 + 4];
            ab.x = is_x ? u0 : fillv;
            ab.y = is_x ? u1 : fillv;
            acc = wmma_acc(ab, acc);
        }
    }

    // Cross-wave reduction of the 16x16 f32 accumulator (8 VGPRs x 32 lanes).
    #pragma unroll
    for (int k = 0; k < 8; ++k)
        lds_part[wave * 256 + k * 32 + lane] = acc[k];
    __syncthreads();

    {
        float s = 0.f;
        #pragma unroll
        for (int w = 0; w < 8; ++w) s += lds_part[w * 256 + tid];
        // Flat slot tid = (vgpr k = tid>>5, lane l = tid&31)
        // -> element r = k + 8*(l>>4), col = l&15 (C/D layout, ISA 7.12.2)
        const int k  = tid >> 5;
        const int l  = tid & 31;
        const int r  = k + 8 * (l >> 4);
        const int cc = l & 15;
        Z[r * 16 + cc] = s;
    }
    __syncthreads();

    // Tiny per-head epilogue: 10 threads, one head each.
    if (tid < NHEAD) {
        const int i = tid;
        float SV0 = 0.f, SV1 = 0.f, s10 = 0.f, s11 = 0.f;
        #pragma unroll
        for (int f = 0; f < 4; ++f) {
            float wvf = 0.f;
            #pragma unroll
            for (int g = 0; g < 4; ++g) wvf += Wv[i * 16 + g * 4 + f];
            SV0 += wvf * Z[8 * 16 + f];       // sum over d of V[s=0]
            SV1 += wvf * Z[8 * 16 + 4 + f];   // sum over d of V[s=1]
            #pragma unroll
            for (int fp = 0; fp < 4; ++fp) {
                float mqk = 0.f;
                #pragma unroll
                for (int g = 0; g < 4; ++g)
                    mqk += Wq[i * 16 + g * 4 + f] * Wk[i * 16 + g * 4 + fp];
                s10 += mqk * Z[(4 + f) * 16 + fp];       // scores[1,0]
                s11 += mqk * Z[(4 + f) * 16 + 4 + fp];   // scores[1,1]
            }
        }
        // Row t=0 softmax is [1,0] under the causal mask -> y0 = SV0.
        const float mx = fmaxf(s10, s11);
        const float e0 = __expf(s10 - mx);
        const float e1 = __expf(s11 - mx);
        const float inv = 1.0f / (e0 + e1);
        // out layout (B, 2, 10): out[b, t, i]
        out[(size_t)b * 2 * NHEAD + i]         = SV0;
        out[(size_t)b * 2 * NHEAD + NHEAD + i] = (e0 * SV0 + e1 * SV1) * inv;
    }
}

extern "C" void kernel_launch(void* const* d_in, const int* in_sizes, int n_in,
                              void* d_out, int out_size, void* d_ws, size_t ws_size,
                              hipStream_t stream) {
    const float* x  = (const float*)d_in[0];
    const float* Wk = (const float*)d_in[1];
    const float* Wq = (const float*)d_in[2];
    const float* Wv = (const float*)d_in[3];
    float* out = (float*)d_out;

    // in_sizes[0] = B * 2B * 2 = 4*B^2  ->  B
    int B = 1;
    {
        long long n = in_sizes[0] / 4;
        while ((long long)(B + 1) * (B + 1) <= n) ++B;   // integer sqrt (B=1024)
    }

    head_attn_gram_wmma<<<dim3(B), dim3(256), 0, stream>>>(x, Wk, Wq, Wv, out, B);
}